// ConvHead_78675210928169
// MI455X (gfx1250) — compile-verified
//
#include <hip/hip_runtime.h>
#include <math.h>

typedef float v2f __attribute__((ext_vector_type(2)));
typedef float v8f __attribute__((ext_vector_type(8)));

#define B_   64
#define C_   256
#define L_   2048
#define H_   8
#define KW   3
#define LP   2046      // L - KW + 1
#define CK   768       // C * KW, GEMM K-dim
#define TOPK 64

// order-preserving float -> uint key (monotone for all finite floats)
__device__ __forceinline__ unsigned sortkey(float x) {
  unsigned u = __float_as_uint(x);
  return (u & 0x80000000u) ? ~u : (u | 0x80000000u);
}

// One block per batch b. 256 threads = 8 wave32s.
// Phase 1: fp32 WMMA conv  XI(16pad x 2046) = W(16pad x 768) x S(768 x 2046)
// Phase 2: per-head BN + top-k threshold + sigmoid mask (wave w -> head w)
// Phase 3: K-shifted sum + head mix -> gate[b, l] to global workspace
extern "C" __global__ void __launch_bounds__(256)
convhead_gate_kernel(const float* __restrict__ src,
                     const float* __restrict__ conv_w,
                     const float* __restrict__ conv_b,
                     const float* __restrict__ bn_gamma,
                     const float* __restrict__ bn_beta,
                     const float* __restrict__ comb_w,
                     float* __restrict__ gate)
{
  extern __shared__ float smem[];
  float* Wlds = smem;           // H_*CK   = 6144 floats (24 KB)
  float* Xl   = smem + H_ * CK; // H_*L_   = 16384 floats (64 KB): xi -> xn -> m
  const int tid  = threadIdx.x;
  const int lane = tid & 31;
  const int wave = tid >> 5;
  const int b    = blockIdx.x;
  const float* srcB = src + (size_t)b * C_ * L_;

  // stage weights: conv_w is already (H, C*K) row-major = GEMM A rows
  for (int i = tid; i < H_ * CK; i += 256) Wlds[i] = conv_w[i];

  // uniform-address preloads (scalarized by the compiler, hoisted out of loops)
  float cbv[H_];
  #pragma unroll
  for (int r = 0; r < H_; ++r) cbv[r] = conv_b[r];

  __syncthreads();

  const int col  = lane & 15;   // A: M index (head, 8..15 = zero pad); B/D: N index
  const int half = lane >> 4;

  // ---- Phase 1: conv via V_WMMA_F32_16X16X4_F32 ----
  for (int t = wave; t < L_ / 16; t += 8) {
    const int l0 = t * 16;
    v8f acc = {0.f, 0.f, 0.f, 0.f, 0.f, 0.f, 0.f, 0.f};
    for (int kk = 0; kk < CK / 4; ++kk) {
      // flat K indices for this lane's .x / .y slots
      // A 16x4 layout (ISA 7.12.2): lanes0-15 -> K{0,1}, lanes16-31 -> K{2,3}
      const int f0 = kk * 4 + half * 2;
      const int f1 = f0 + 1;
      // A operand: weights, rows >= H_ are zero padding
      const int wr = (col < H_) ? col : (H_ - 1);
      float ax = Wlds[wr * CK + f0];
      float ay = Wlds[wr * CK + f1];
      ax = (col < H_) ? ax : 0.f;
      ay = (col < H_) ? ay : 0.f;
      // B operand: S[f][l0+col] = src[c(f), l0+col+j(f)], f = c*3 + j
      // (4x16 B layout assumed to mirror A: VGPR0 rows K0|K2, VGPR1 rows K1|K3)
      const int c0 = f0 / 3, j0 = f0 - c0 * 3;
      const int c1 = f1 / 3, j1 = f1 - c1 * 3;
      int i0 = l0 + col + j0; i0 = (i0 < L_) ? i0 : (L_ - 1); // clamp: tail cols discarded
      int i1 = l0 + col + j1; i1 = (i1 < L_) ? i1 : (L_ - 1);
      float bx = srcB[c0 * L_ + i0];
      float by = srcB[c1 * L_ + i1];
      v2f av = {ax, ay};
      v2f bv = {bx, by};
      acc = __builtin_amdgcn_wmma_f32_16x16x4_f32(false, av, false, bv,
                                                  (short)0, acc, false, false);
    }
    // D layout: VGPR r: lanes0-15 -> M=r, lanes16-31 -> M=r+8; N = col.
    // Only lanes 0-15 (half==0) hold real heads (M < 8): one predicate, 8 stores.
    const int l = l0 + col;
    if (half == 0 && l < LP) {
      #pragma unroll
      for (int r = 0; r < 8; ++r)
        Xl[r * L_ + l] = acc[r] + cbv[r];
    }
  }
  __syncthreads();

  // ---- Phase 2: per-head BN + top-k + soft-mask (wave w owns head w) ----
  {
    const int h = wave;
    float s = 0.f, ss = 0.f;
    for (int l = lane; l < LP; l += 32) {
      float x = Xl[h * L_ + l];
      s += x; ss += x * x;
    }
    for (int o = 16; o > 0; o >>= 1) {
      s  += __shfl_xor(s,  o, 32);
      ss += __shfl_xor(ss, o, 32);
    }
    const float inv = 1.f / (float)LP;
    const float mu  = s * inv;
    const float var = ss * inv - mu * mu;          // biased var, as in BN train mode
    const float scale = bn_gamma[h] * rsqrtf(var + 1e-5f);
    const float shift = bn_beta[h] - mu * scale;
    for (int l = lane; l < LP; l += 32)
      Xl[h * L_ + l] = Xl[h * L_ + l] * scale + shift;

    // k-th largest value via 32-step radix search on monotone keys
    unsigned kth = 0u;
    for (int bit = 31; bit >= 0; --bit) {
      const unsigned cand = kth | (1u << bit);
      int cnt = 0;
      for (int l = lane; l < LP; l += 32)
        cnt += (sortkey(Xl[h * L_ + l]) >= cand) ? 1 : 0;
      for (int o = 16; o > 0; o >>= 1) cnt += __shfl_xor(cnt, o, 32);
      if (cnt >= TOPK) kth = cand;                 // predicate monotone in cand
    }

    // m = sigmoid(xn) * [xn >= kth]; zero the [LP, L_) tail for the shift-sum
    for (int l = lane; l < L_; l += 32) {
      float m = 0.f;
      if (l < LP) {
        const float x = Xl[h * L_ + l];
        if (sortkey(x) >= kth) m = 1.f / (1.f + expf(-x));
      }
      Xl[h * L_ + l] = m;
    }
  }
  __syncthreads();

  // ---- Phase 3: gate[b,l] = (1/KW) * sum_h comb_w[h] * sum_i m[h, l-i] ----
  float cw[H_];
  #pragma unroll
  for (int r = 0; r < H_; ++r) cw[r] = comb_w[r];

  for (int l = tid; l < L_; l += 256) {
    float g = 0.f;
    #pragma unroll
    for (int h = 0; h < H_; ++h) {
      float v = Xl[h * L_ + l];
      if (l >= 1) v += Xl[h * L_ + l - 1];
      if (l >= 2) v += Xl[h * L_ + l - 2];
      g += cw[h] * v;
    }
    gate[b * L_ + l] = g * (1.f / (float)KW);
  }
}

// out[b,c,l] = src[b,c,l] * gate[b,l] + comb_b  — streaming, float4, L2-friendly
extern "C" __global__ void __launch_bounds__(256)
convhead_apply_kernel(const float* __restrict__ src,
                      const float* __restrict__ gate,
                      const float* __restrict__ comb_b,
                      float* __restrict__ out)
{
  const float cb = comb_b[0];
  const size_t total = (size_t)B_ * C_ * (L_ / 4);        // 8,388,608 float4
  const float4* s4 = (const float4*)src;
  const float4* g4 = (const float4*)gate;
  float4*       o4 = (float4*)out;
  for (size_t v = (size_t)blockIdx.x * blockDim.x + threadIdx.x; v < total;
       v += (size_t)gridDim.x * blockDim.x) {
    const int p = (int)(v & (L_ / 4 - 1));                // float4 pos in row
    const int b = (int)(v >> 17);                         // / (C_ * L_/4)
    const float4 s = s4[v];
    const float4 g = g4[((size_t)b << 9) | p];
    float4 r;
    r.x = s.x * g.x + cb;
    r.y = s.y * g.y + cb;
    r.z = s.z * g.z + cb;
    r.w = s.w * g.w + cb;
    o4[v] = r;
  }
}

extern "C" void kernel_launch(void* const* d_in, const int* in_sizes, int n_in,
                              void* d_out, int out_size, void* d_ws, size_t ws_size,
                              hipStream_t stream) {
  (void)in_sizes; (void)n_in; (void)out_size; (void)ws_size;
  const float* src      = (const float*)d_in[0];
  const float* conv_w   = (const float*)d_in[1];
  const float* conv_b   = (const float*)d_in[2];
  const float* bn_gamma = (const float*)d_in[3];
  const float* bn_beta  = (const float*)d_in[4];
  const float* comb_w   = (const float*)d_in[5];
  const float* comb_b   = (const float*)d_in[6];
  float* out  = (float*)d_out;
  float* gate = (float*)d_ws;                      // B_*L_ floats = 512 KB

  const size_t lds_bytes = (size_t)(H_ * CK + H_ * L_) * sizeof(float); // 88 KB
  convhead_gate_kernel<<<dim3(B_), dim3(256), lds_bytes, stream>>>(
      src, conv_w, conv_b, bn_gamma, bn_beta, comb_w, gate);
  convhead_apply_kernel<<<dim3(8192), dim3(256), 0, stream>>>(
      src, gate, comb_b, out);
}